// MPJPE_Loss_78735340470496
// MI455X (gfx1250) — compile-verified
//
#include <hip/hip_runtime.h>
#include <hip/hip_bf16.h>
#include <math.h>

#define NUM_JOINTS 17
#define NUM_PAIRS  14
#define GRID_N     64
#define BATCH      256
#define NCH        (NUM_JOINTS + 3 * NUM_PAIRS)   // 59

typedef __attribute__((ext_vector_type(2))) float v2f;
typedef __attribute__((ext_vector_type(8))) float v8f;

__device__ __constant__ int PARENT_P[NUM_PAIRS] = {0,1,2,0,4,5,0,8,14,15,8,11,12,8};
__device__ __constant__ int PARENT_C[NUM_PAIRS] = {1,2,3,4,5,6,8,14,15,16,11,12,13,10};

// ---------------- workspace layout (floats) ----------------
// gx      : [B][17][64]            offset 0
// gy      : [B][17][64]            offset 278528
// pairSums: [B][14][3]             offset 557056
// s2d     : [1]                    offset 567808
#define WS_GX   0
#define WS_GY   278528
#define WS_PS   557056
#define WS_S2D  567808

__device__ inline float waveSum(float v) {
    #pragma unroll
    for (int off = 16; off; off >>= 1) v += __shfl_xor(v, off, 32);
    return v;
}
__device__ inline float waveMin(float v) {
    #pragma unroll
    for (int off = 16; off; off >>= 1) v = fminf(v, __shfl_xor(v, off, 32));
    return v;
}

// Zero the atomic accumulator(s). (Workspace is poisoned once with 0xAA and
// never re-poisoned; we must re-zero every launch.)
__global__ void zero_kernel(float* ws) {
    ws[WS_S2D] = 0.0f;
}

// One block (64 threads) per (b, joint): normalized separable 1-D Gaussian
// profiles. heatmap(b,j,i,k) = gx[i]*gy[k] exactly (max-normalization factors).
__global__ void profile_kernel(const float* __restrict__ joint2d,
                               float* __restrict__ ws) {
    const int bj = blockIdx.x;          // b*17 + j
    const int t  = threadIdx.x;         // 0..63
    const float x = joint2d[bj * 2 + 0] * 64.0f;
    const float y = joint2d[bj * 2 + 1] * 64.0f;
    const float dx = (float)t - x, dy = (float)t - y;
    const float dx2 = dx * dx, dy2 = dy * dy;

    __shared__ float smx[2], smy[2];
    float mx = waveMin(dx2);
    float my = waveMin(dy2);
    const int wid = t >> 5;
    if ((t & 31) == 0) { smx[wid] = mx; smy[wid] = my; }
    __syncthreads();
    mx = fminf(smx[0], smx[1]);
    my = fminf(smy[0], smy[1]);

    ws[WS_GX + bj * 64 + t] = __expf(-(dx2 - mx) * 0.25f);
    ws[WS_GY + bj * 64 + t] = __expf(-(dy2 - my) * 0.25f);
}

// One block per (b, channel). 256 threads read the 64x64 tile once (float4 /
// b128 loads, fully coalesced), form target = alpha*hp + beta*hc from LDS-cached
// 1-D profiles, square the diff, and reduce with V_WMMA_F32_16X16X4_F32 using
// an all-ones A matrix (exact f32 column-sum reduction).
__global__ void main_kernel(const float* __restrict__ middle,
                            const float* __restrict__ joints,
                            float* __restrict__ ws) {
    const int blk = blockIdx.x;
    const int b   = blk / NCH;
    const int c   = blk - b * NCH;
    const int t   = threadIdx.x;

    __shared__ float sgx0[64], sgy0[64], sgx1[64], sgy1[64];
    __shared__ float sred[8];

    float alpha, beta;
    int j0, j1;
    if (c < NUM_JOINTS) {
        j0 = c; j1 = c;                 // duplicate; beta = 0
        alpha = 1.0f; beta = 0.0f;
    } else {
        const int q  = c - NUM_JOINTS;
        const int p  = q / 3;
        const int ch = q - p * 3;
        j0 = PARENT_P[p];
        j1 = PARENT_C[p];
        const float zp = joints[(b * NUM_JOINTS + j0) * 3 + 2];
        const float zc = joints[(b * NUM_JOINTS + j1) * 3 + 2];
        const float d  = zp - zc;
        const int r = (d > 0.1f) ? 1 : ((fabsf(d) < 0.1f) ? 0 : -1);
        if (ch == 0)      { alpha = 0.0f; beta = (r == -1) ? 1.0f : 0.0f; }
        else if (ch == 1) { alpha = 1.0f; beta = (r ==  0) ? 1.0f : 0.0f; }
        else              { alpha = 0.0f; beta = (r ==  1) ? 1.0f : 0.0f; }
    }

    // Stage the (at most) two 1-D profile pairs into LDS.
    {
        const float* gxb = ws + WS_GX + (size_t)b * NUM_JOINTS * 64;
        const float* gyb = ws + WS_GY + (size_t)b * NUM_JOINTS * 64;
        if (t < 64)       sgx0[t]       = gxb[j0 * 64 + t];
        else if (t < 128) sgy0[t - 64]  = gyb[j0 * 64 + (t - 64)];
        else if (t < 192) sgx1[t - 128] = gxb[j1 * 64 + (t - 128)];
        else              sgy1[t - 192] = gyb[j1 * 64 + (t - 192)];
    }
    __syncthreads();

    const float* base = middle + ((size_t)b * NCH + c) * (GRID_N * GRID_N);

    v8f Cacc = {};
    v2f Aones; Aones.x = 1.0f; Aones.y = 1.0f;

    #pragma unroll
    for (int rep = 0; rep < 4; ++rep) {
        const int n  = rep * 1024 + t * 4;       // linear pixel index, 16B aligned
        const int i  = n >> 6;                   // row (dx axis)
        const int k0 = n & 63;                   // col (dy axis), multiple of 4
        const float4 m = *reinterpret_cast<const float4*>(base + n);

        const float tx0 = alpha * sgx0[i];
        const float tx1 = beta  * sgx1[i];
        const float d0 = tx0 * sgy0[k0 + 0] + tx1 * sgy1[k0 + 0] - m.x;
        const float d1 = tx0 * sgy0[k0 + 1] + tx1 * sgy1[k0 + 1] - m.y;
        const float d2 = tx0 * sgy0[k0 + 2] + tx1 * sgy1[k0 + 2] - m.z;
        const float d3 = tx0 * sgy0[k0 + 3] + tx1 * sgy1[k0 + 3] - m.w;

        v2f Bv; Bv.x = d0 * d0 + d1 * d1;
                Bv.y = d2 * d2 + d3 * d3;

        // D = ones(16x4) x B(4x16) + C : every row of D is the column sums of B.
        Cacc = __builtin_amdgcn_wmma_f32_16x16x4_f32(
            false, Aones, false, Bv, (short)0, Cacc, false, false);
    }

    // C VGPR0: lanes 0-15 hold colsums N=0..15 (row M=0); lanes 16-31 hold the
    // identical row M=8 -> 32-lane sum double-counts; halve it.
    float waveTotal = 0.5f * waveSum(Cacc[0]);

    const int wid = t >> 5;
    if ((t & 31) == 0) sred[wid] = waveTotal;
    __syncthreads();

    if (t == 0) {
        float blockTotal = 0.0f;
        #pragma unroll
        for (int w = 0; w < 8; ++w) blockTotal += sred[w];
        if (c < NUM_JOINTS) {
            atomicAdd(ws + WS_S2D, blockTotal);
        } else {
            const int q  = c - NUM_JOINTS;
            const int p  = q / 3;
            const int ch = q - p * 3;
            ws[WS_PS + ((size_t)b * NUM_PAIRS + p) * 3 + ch] = blockTotal;
        }
    }
}

// One block: l3d (L1 over 13056 elems), LHEM combine over 3584 (b,p) entries,
// final scalar.
__global__ void final_kernel(const float* __restrict__ pred,
                             const float* __restrict__ joints,
                             const float* __restrict__ ws,
                             float* __restrict__ out) {
    const int t = threadIdx.x;

    float l3 = 0.0f;
    for (int idx = t; idx < BATCH * NUM_JOINTS * 3; idx += 256)
        l3 += fabsf(joints[idx] - pred[idx]);

    float lh = 0.0f;
    for (int e = t; e < BATCH * NUM_PAIRS; e += 256) {
        const float* ps = ws + WS_PS + (size_t)e * 3;
        const float s = sqrtf(ps[0]) + sqrtf(ps[1]) + sqrtf(ps[2]);
        lh += s * s;
    }

    __shared__ float red[16];
    l3 = waveSum(l3);
    lh = waveSum(lh);
    const int wid = t >> 5;
    if ((t & 31) == 0) { red[wid] = l3; red[8 + wid] = lh; }
    __syncthreads();

    if (t == 0) {
        float L3 = 0.0f, LH = 0.0f;
        #pragma unroll
        for (int w = 0; w < 8; ++w) { L3 += red[w]; LH += red[8 + w]; }
        const float invB = 1.0f / (float)BATCH;
        out[0] = L3 * invB + 0.005f * (LH * invB + ws[WS_S2D] * invB);
    }
}

extern "C" void kernel_launch(void* const* d_in, const int* in_sizes, int n_in,
                              void* d_out, int out_size, void* d_ws, size_t ws_size,
                              hipStream_t stream) {
    const float* pred    = (const float*)d_in[0];
    const float* joints  = (const float*)d_in[1];
    const float* middle  = (const float*)d_in[2];
    const float* joint2d = (const float*)d_in[3];
    float* ws  = (float*)d_ws;
    float* out = (float*)d_out;

    zero_kernel<<<1, 1, 0, stream>>>(ws);
    profile_kernel<<<BATCH * NUM_JOINTS, 64, 0, stream>>>(joint2d, ws);
    main_kernel<<<BATCH * NCH, 256, 0, stream>>>(middle, joints, ws);
    final_kernel<<<1, 256, 0, stream>>>(pred, joints, ws, out);
}